// SelectiveSSM_88734024336016
// MI455X (gfx1250) — compile-verified
//
#include <hip/hip_runtime.h>
#include <cstdint>

#define BATCH   2
#define SEQ_L   1024
#define DMODEL  1024
#define DSTATE  16
#define DCONV   4
#define DINNER  2048
#define DTRANK  64
#define MROWS   (BATCH * SEQ_L)       // 2048
#define SSMW    (DTRANK + 2 * DSTATE) // 96 (true width)
#define SSMP    128                   // padded row stride for ssm / W_xp

typedef float v2f __attribute__((ext_vector_type(2)));
typedef float v8f __attribute__((ext_vector_type(8)));

// ---------------------------------------------------------------------------
// Branch-free row-major f32 GEMM: C[M,N] = A[M,K](lda) * B[K,N](ldb), ldc.
// Requires M%128==0, N%128==0, K%16==0 (guaranteed by padding).
// Block = 256 threads = 8 waves arranged 4(M) x 2(N); block tile 128x128.
// Wave tile 32x64 = 2x4 WMMA 16x16 tiles via V_WMMA_F32_16X16X4_F32.
// B tile staged into LDS with GLOBAL_LOAD_ASYNC_TO_LDS_B128 (ASYNCcnt path).
// NOTE: &Bs must escape into the asm so the compiler knows the DMA writes it
// (otherwise loads of a store-less internal LDS global fold to undef).
// ---------------------------------------------------------------------------
__global__ __launch_bounds__(256) void wmma_gemm_f32(
    const float* __restrict__ A, const float* __restrict__ B,
    float* __restrict__ C, int M, int N, int K, int lda, int ldb, int ldc) {
  __shared__ float Bs[16][132];            // 132 pad: 16B-aligned rows, no bank alias
  const int tid   = threadIdx.x;
  const int wave  = tid >> 5;              // 0..7
  const int lane  = tid & 31;
  const int waveM = wave >> 1;             // 0..3
  const int waveN = wave & 1;              // 0..1
  const int m0    = blockIdx.x * 128 + waveM * 32;
  const int nblk  = blockIdx.y * 128;
  const int n0w   = waveN * 64;            // col offset inside block tile
  const int lrow  = lane & 15;
  const int khalf = (lane >> 4) * 2;       // lanes 16-31 hold K pair +2,+3

  // B staging map: thread t -> row sr=t/16 (0..15), cols sc=(t%16)*8 .. +7
  const int sr = tid >> 4;
  const int sc = (tid & 15) * 8;
  const unsigned ldsOff = (unsigned)((sr * 132 + sc) * 4);  // Bs at LDS base 0

  v8f acc[2][4] = {};

  for (int k0 = 0; k0 < K; k0 += 16) {
    const uint64_t gaddr = (uint64_t)(B + (size_t)(k0 + sr) * ldb + nblk + sc);
    __builtin_prefetch((const float*)gaddr + (size_t)16 * ldb, 0, 3); // next slice
    __syncthreads();                       // previous slice fully consumed
    // 32B per thread, memory -> LDS via async DMA path (tracked by ASYNCcnt);
    // offset:16 applies to both the LDS and the global side (ISA 08 §4.4).
    // %2 (address of Bs) is unreferenced but escapes the LDS object so the
    // compiler treats it as written by this asm ("memory" clobber).
    asm volatile(
        "global_load_async_to_lds_b128 %0, %1, off\n\t"
        "global_load_async_to_lds_b128 %0, %1, off offset:16\n\t"
        "s_wait_asynccnt 0x0"
        :
        : "v"(ldsOff), "v"(gaddr), "v"((const float*)&Bs[0][0])
        : "memory");
    __syncthreads();                       // all waves' async stores visible

#pragma unroll
    for (int ks = 0; ks < 4; ++ks) {
      const int kk = k0 + ks * 4;
      v2f a[2];
#pragma unroll
      for (int i = 0; i < 2; ++i) {
        const float* ap = A + (size_t)(m0 + i * 16 + lrow) * lda + kk + khalf;
        a[i].x = ap[0];                    // merged to global_load_b64
        a[i].y = ap[1];
      }
      v2f b[4];
#pragma unroll
      for (int j = 0; j < 4; ++j) {
        const int col = n0w + j * 16 + lrow;
        b[j].x = Bs[ks * 4 + khalf][col];
        b[j].y = Bs[ks * 4 + khalf + 1][col];
      }
#pragma unroll
      for (int i = 0; i < 2; ++i)
#pragma unroll
        for (int j = 0; j < 4; ++j)
          acc[i][j] = __builtin_amdgcn_wmma_f32_16x16x4_f32(
              false, a[i], false, b[j], (short)0, acc[i][j], false, false);
    }
  }

  const int rhalf = (lane >> 4) * 8;       // C/D: lanes 16-31 hold rows e+8
#pragma unroll
  for (int i = 0; i < 2; ++i)
#pragma unroll
    for (int j = 0; j < 4; ++j)
#pragma unroll
      for (int e = 0; e < 8; ++e)
        C[(size_t)(m0 + i * 16 + e + rhalf) * ldc + nblk + n0w + j * 16 + lrow] =
            acc[i][j][e];
}

// ---------------------------------------------------------------------------
// Zero-pad W_x (DINNER x 96) -> W_xp (DINNER x 128).
// ---------------------------------------------------------------------------
__global__ __launch_bounds__(256) void pad_wx_kernel(
    const float* __restrict__ W_x, float* __restrict__ W_xp) {
  const int idx = blockIdx.x * 256 + threadIdx.x;   // DINNER*128 total
  if (idx >= DINNER * SSMP) return;
  const int col = idx & (SSMP - 1);
  const int row = idx >> 7;
  W_xp[idx] = (col < SSMW) ? W_x[row * SSMW + col] : 0.f;
}

// ---------------------------------------------------------------------------
// Causal depthwise conv (k=4, left pad 3) + bias + SiLU.
// ---------------------------------------------------------------------------
__global__ __launch_bounds__(256) void conv_silu_kernel(
    const float* __restrict__ xv0, const float* __restrict__ w,
    const float* __restrict__ bias, float* __restrict__ xv) {
  const size_t idx = (size_t)blockIdx.x * blockDim.x + threadIdx.x;
  if (idx >= (size_t)MROWS * DINNER) return;
  const int d     = (int)(idx % DINNER);
  const size_t bl = idx / DINNER;
  const int l     = (int)(bl % SEQ_L);
  const size_t brow = bl - (size_t)l;     // b*L
  float acc = bias[d];
#pragma unroll
  for (int j = 0; j < DCONV; ++j) {
    const int ls = l - (DCONV - 1) + j;
    if (ls >= 0) acc += xv0[(brow + ls) * DINNER + d] * w[d * DCONV + j];
  }
  xv[idx] = acc / (1.f + __expf(-acc));   // silu
}

// ---------------------------------------------------------------------------
// delta = softplus(deltaRaw + b_dt), in place.
// ---------------------------------------------------------------------------
__global__ __launch_bounds__(256) void softplus_bias_kernel(
    const float* __restrict__ din, const float* __restrict__ b_dt,
    float* __restrict__ dout) {
  const size_t idx = (size_t)blockIdx.x * blockDim.x + threadIdx.x;
  if (idx >= (size_t)MROWS * DINNER) return;
  const int d = (int)(idx % DINNER);
  const float v = din[idx] + b_dt[d];
  dout[idx] = (v > 20.f) ? v : log1pf(__expf(v));
}

// ---------------------------------------------------------------------------
// Selective scan. One lane per (d, n): lanes 0-15 -> channel d0, lanes 16-31
// -> channel d0+1; wave walks L sequentially, shuffle-xor reduces 16 states.
// ssm rows have stride SSMP (=128): B at col 64+n, C at col 80+n.
// ---------------------------------------------------------------------------
__global__ __launch_bounds__(256) void scan_kernel(
    const float* __restrict__ delta, const float* __restrict__ xv,
    const float* __restrict__ ssm,   const float* __restrict__ A_log,
    const float* __restrict__ Dvec,  float* __restrict__ y) {
  const int wave  = threadIdx.x >> 5;
  const int lane  = threadIdx.x & 31;
  const int gw    = blockIdx.x * 8 + wave;          // 0..2047
  const int b     = gw >> 10;
  const int dbase = (gw & 1023) * 2;
  const int d     = dbase + (lane >> 4);
  const int n     = lane & 15;

  const float Aval = -__expf(A_log[d * DSTATE + n]);
  const float Dd   = Dvec[d];
  float h = 0.f;

  const size_t browD = (size_t)b * SEQ_L * DINNER;
  const size_t browS = (size_t)b * SEQ_L * SSMP;

  for (int l = 0; l < SEQ_L; ++l) {
    const size_t bld = browD + (size_t)l * DINNER + d;
    const size_t bls = browS + (size_t)l * SSMP;
    const float dlt = delta[bld];
    const float xvl = xv[bld];
    const float Bn  = ssm[bls + DTRANK + n];
    const float Cn  = ssm[bls + DTRANK + DSTATE + n];
    __builtin_prefetch(delta + bld + DINNER, 0, 3);  // speculative, branchless
    __builtin_prefetch(ssm + bls + SSMP, 0, 3);
    const float dA = __expf(dlt * Aval);
    h = dA * h + dlt * Bn * xvl;
    float contrib = h * Cn;
#pragma unroll
    for (int m = 1; m < DSTATE; m <<= 1)
      contrib += __shfl_xor(contrib, m, 32);         // reduce 16-lane group
    if (n == 0) y[bld] = contrib + xvl * Dd;
  }
}

// ---------------------------------------------------------------------------
extern "C" void kernel_launch(void* const* d_in, const int* in_sizes, int n_in,
                              void* d_out, int out_size, void* d_ws, size_t ws_size,
                              hipStream_t stream) {
  const float* x      = (const float*)d_in[0];
  const float* W_in   = (const float*)d_in[1];
  const float* conv_w = (const float*)d_in[2];
  const float* conv_b = (const float*)d_in[3];
  const float* W_x    = (const float*)d_in[4];
  const float* W_dt   = (const float*)d_in[5];
  const float* b_dt   = (const float*)d_in[6];
  const float* A_log  = (const float*)d_in[7];
  const float* Dvec   = (const float*)d_in[8];
  const float* W_out  = (const float*)d_in[9];
  float* out = (float*)d_out;

  const size_t NE = (size_t)MROWS * DINNER;          // 4M elements
  float* ws    = (float*)d_ws;
  float* xv0   = ws;                                 // [MROWS, DINNER]; reused as y
  float* xv    = xv0 + NE;                           // [MROWS, DINNER]
  float* W_xp  = xv + NE;                            // [DINNER, 128] padded
  float* ssm   = W_xp + (size_t)DINNER * SSMP;       // [MROWS, 128] padded
  float* delta = ssm + (size_t)MROWS * SSMP;         // [MROWS, DINNER]
  float* y     = xv0;                                // alias: xv0 dead after conv

  const int elemBlocks = (int)((NE + 255) / 256);

  // 0) pad W_x to 128-wide
  pad_wx_kernel<<<(DINNER * SSMP + 255) / 256, 256, 0, stream>>>(W_x, W_xp);

  // 1) xv0 = x @ W_in            (2048 x 1024) * (1024 x 2048)
  wmma_gemm_f32<<<dim3(MROWS / 128, DINNER / 128), 256, 0, stream>>>(
      x, W_in, xv0, MROWS, DINNER, DMODEL, DMODEL, DINNER, DINNER);

  // 2) depthwise conv + bias + silu
  conv_silu_kernel<<<elemBlocks, 256, 0, stream>>>(xv0, conv_w, conv_b, xv);

  // 3) ssm = xv @ W_xp           (2048 x 2048) * (2048 x 128)
  wmma_gemm_f32<<<dim3(MROWS / 128, SSMP / 128), 256, 0, stream>>>(
      xv, W_xp, ssm, MROWS, SSMP, DINNER, DINNER, SSMP, SSMP);

  // 4) deltaRaw = ssm[:, :64] @ W_dt   (2048 x 64) * (64 x 2048), lda=128
  wmma_gemm_f32<<<dim3(MROWS / 128, DINNER / 128), 256, 0, stream>>>(
      ssm, W_dt, delta, MROWS, DINNER, DTRANK, SSMP, DINNER, DINNER);

  // 5) delta = softplus(deltaRaw + b_dt)
  softplus_bias_kernel<<<elemBlocks, 256, 0, stream>>>(delta, b_dt, delta);

  // 6) selective scan -> y (plus skip term xv * D)
  scan_kernel<<<(BATCH * DINNER / 2) / 8, 256, 0, stream>>>(
      delta, xv, ssm, A_log, Dvec, y);

  // 7) out = y @ W_out           (2048 x 2048) * (2048 x 1024)
  wmma_gemm_f32<<<dim3(MROWS / 128, DMODEL / 128), 256, 0, stream>>>(
      y, W_out, out, MROWS, DMODEL, DINNER, DINNER, DMODEL, DMODEL);
}